// AntiAliasActivation_601295421761
// MI455X (gfx1250) — compile-verified
//
#include <hip/hip_runtime.h>
#include <cstdint>

// AntiAliasActivation fused kernel for gfx1250 (MI455X).
// Single pass: upsample(x2,12tap) -> snake_beta -> downsample(x2,12tap).
// Memory-bound: 268MB min traffic @23.3TB/s ~= 11.5us. Fusion avoids the 2x
// intermediate. Data path: async global->LDS b128 staging (ASYNCcnt),
// ds_load_b128 register windows, non-temporal b128 output stores.

#define BB      8
#define CC      512
#define TT      8192
#define CHUNK   2048
#define THREADS 256
#define TPT     8
#define NCHUNK  (TT / CHUNK)   // 4 chunks per row

static_assert(NCHUNK == 4, "row split assumed to be 4 chunks");
static_assert(CHUNK == THREADS * TPT, "one chunk per block");

typedef __attribute__((ext_vector_type(4))) float v4f;  // native vector for b128 ops

__device__ __forceinline__ float snake_f(float y, float a, float binv) {
    float s = __sinf(y * a);
    return fmaf(s * s, binv, y);
}

// gfx1250 async copy global -> LDS, 16B per lane, tracked by ASYNCcnt.
__device__ __forceinline__ void async_b128_to_lds(uint32_t lds_byte_off,
                                                  const float* gaddr) {
    asm volatile("global_load_async_to_lds_b128 %0, %1, off"
                 :: "v"(lds_byte_off), "v"(gaddr)
                 : "memory");
}

__global__ __launch_bounds__(THREADS)
void aa_snake_fused_kernel(const float* __restrict__ x,
                           const float* __restrict__ alpha,
                           const float* __restrict__ beta,
                           const float* __restrict__ upf,
                           const float* __restrict__ dnf,
                           float* __restrict__ out) {
    // LDS tile: x[cs-8 .. cs+CHUNK+8) for one (b,c) row chunk.
    __shared__ __align__(16) float sx[CHUNK + 16];

    const int tid     = threadIdx.x;
    const int bid     = blockIdx.x;
    const int chunk_i = bid & (NCHUNK - 1);
    const int row     = bid >> 2;               // row = b*C + c
    const int c       = row & (CC - 1);
    const size_t roff = (size_t)row * TT;
    const float* xrow = x + roff;
    const int cs      = chunk_i * CHUNK;

    // ---- Stage input tile: bulk via async b128 (512 transfers = 2/thread) ----
    // sx[8+e] = xrow[cs+e], e in [0, CHUNK)
    #pragma unroll
    for (int rI = 0; rI < 2; ++rI) {
        int e = (rI * THREADS + tid) * 4;
        uint32_t loff = (uint32_t)(uintptr_t)(&sx[8 + e]);   // low32 = LDS offset
        async_b128_to_lds(loff, xrow + cs + e);
    }
    // Halos (8 left + 8 right), edge-clamped scalar loads.
    if (tid < 16) {
        int li = (tid < 8) ? tid : (CHUNK + tid);            // 0..7 | CHUNK+8..+15
        int g  = (tid < 8) ? (cs - 8 + tid) : (cs + CHUNK + (tid - 8));
        g = min(max(g, 0), TT - 1);
        sx[li] = xrow[g];
    }
    asm volatile("s_wait_asynccnt 0x0" ::: "memory");
    __syncthreads();

    // ---- Per-channel snake parameters (uniform per block) ----
    const float a    = __expf(alpha[c]);
    const float binv = 1.0f / (__expf(beta[c]) + 1e-9f);

    // Polyphase filters (ratio=2 folded in): even t' uses odd taps, odd t' even taps.
    float fe[6], fo[6], gd[12];
    #pragma unroll
    for (int r = 0; r < 6; ++r) {
        fe[r] = 2.0f * upf[2 * r];
        fo[r] = 2.0f * upf[2 * r + 1];
    }
    #pragma unroll
    for (int j = 0; j < 12; ++j) gd[j] = dnf[j];

    // ---- Register x window: xr[i] = x[t0-8+i], i in [0,24), t0 = cs + tid*8 ----
    float xr[24];
    const int base = tid * TPT;                 // LDS index of x[t0-8]
    #pragma unroll
    for (int v = 0; v < 6; ++v) {
        v4f t = *(const v4f*)&sx[base + 4 * v];              // ds_load_b128
        xr[4 * v + 0] = t.x; xr[4 * v + 1] = t.y;
        xr[4 * v + 2] = t.z; xr[4 * v + 3] = t.w;
    }

    const bool firstc = (cs == 0);
    const bool lastc  = (cs + CHUNK == TT);
    // x-level replicate-pad: only the outermost thread of edge chunks needs it.
    if (firstc && tid == 0) {
        #pragma unroll
        for (int i = 0; i < 8; ++i) xr[i] = xr[8];           // x[<0] -> x[0]
    }
    if (lastc && tid == THREADS - 1) {
        #pragma unroll
        for (int i = 16; i < 24; ++i) xr[i] = xr[15];        // x[>T-1] -> x[T-1]
    }

    // ---- Upsample + snake: s_arr[q] = snake(y_up(2*t0-5+q)), q in [0,26) ----
    // q=2k   -> t' = 2*(t0-3+k)+1 (odd t', even taps fe)
    // q=2k+1 -> t' = 2*(t0-2+k)   (even t', odd taps fo)
    // Both draw from the same x window xr[k+3 .. k+8].
    float s_arr[26];
    #pragma unroll
    for (int k = 0; k < 13; ++k) {
        float yo = 0.0f, ye = 0.0f;
        #pragma unroll
        for (int r = 0; r < 6; ++r) {
            float xv = xr[k + 8 - r];
            yo = fmaf(fe[r], xv, yo);
            ye = fmaf(fo[r], xv, ye);
        }
        s_arr[2 * k]     = snake_f(yo, a, binv);
        s_arr[2 * k + 1] = snake_f(ye, a, binv);
    }

    // s-level replicate-pad of the downsample input (clamp t' to [0, 2T-1]).
    if (firstc && tid == 0) {
        #pragma unroll
        for (int q = 0; q < 5; ++q) s_arr[q] = s_arr[5];     // t'<0 -> s(0)
    }
    if (lastc && tid == THREADS - 1) {
        #pragma unroll
        for (int q = 21; q < 26; ++q) s_arr[q] = s_arr[20];  // t'>2T-1 -> s(2T-1)
    }

    // ---- Downsample: y[t0+i] = sum_j gd[j] * s_arr[2i+j] ----
    float acc[TPT];
    #pragma unroll
    for (int i = 0; i < TPT; ++i) {
        float v = 0.0f;
        #pragma unroll
        for (int j = 0; j < 12; ++j) v = fmaf(gd[j], s_arr[2 * i + j], v);
        acc[i] = v;
    }

    // ---- Streaming store (write-once output: keep it out of L2) ----
    float* orow = out + roff + cs + tid * TPT;
    v4f o0 = {acc[0], acc[1], acc[2], acc[3]};
    v4f o1 = {acc[4], acc[5], acc[6], acc[7]};
    __builtin_nontemporal_store(o0, (v4f*)orow);             // global_store_b128 nt
    __builtin_nontemporal_store(o1, (v4f*)(orow + 4));
}

extern "C" void kernel_launch(void* const* d_in, const int* in_sizes, int n_in,
                              void* d_out, int out_size, void* d_ws, size_t ws_size,
                              hipStream_t stream) {
    const float* x     = (const float*)d_in[0];
    const float* alpha = (const float*)d_in[1];
    const float* beta  = (const float*)d_in[2];
    const float* upf   = (const float*)d_in[3];
    const float* dnf   = (const float*)d_in[4];
    float* out         = (float*)d_out;

    const int blocks = BB * CC * NCHUNK;   // 16384 blocks x 256 threads (8 waves)
    aa_snake_fused_kernel<<<dim3(blocks), dim3(THREADS), 0, stream>>>(
        x, alpha, beta, upf, dnf, out);
}